// PIGNODE_6897717477532
// MI455X (gfx1250) — compile-verified
//
#include <hip/hip_runtime.h>
#include <hip/hip_bf16.h>

typedef __attribute__((ext_vector_type(2))) float v2f;
typedef __attribute__((ext_vector_type(8))) float v8f;
typedef __attribute__((ext_vector_type(4))) int   v4i;

#define GRID_W 64
#define NNODE  4096            // GRID_W*GRID_W
#define BATCH  8
#define BNTOT  32768           // BATCH*NNODE
#define IN_DIM 12
#define HID    64
#define HEADS  4
#define FDIM   256             // HEADS*HID
#define DT     0.25f

#if __has_builtin(__builtin_amdgcn_global_load_async_to_lds_b128) && \
    __has_builtin(__builtin_amdgcn_s_wait_asynccnt)
#define HAVE_ASYNC_LDS 1
typedef __attribute__((address_space(1))) v4i gv4i;   // global 16-byte vector
typedef __attribute__((address_space(3))) v4i lv4i;   // LDS    16-byte vector
#else
#define HAVE_ASYNC_LDS 0
#endif

// ---------------------------------------------------------------------------
// a_edge collapses to a 3x4 matrix: v[k][head] = sum_d W_edge[k,h*64+d]*att_edge[h,d]
__global__ void k_edge_v(const float* __restrict__ W_edge,
                         const float* __restrict__ att_edge,
                         float* __restrict__ v) {
    int t = threadIdx.x;
    if (t >= 12) return;
    int k = t >> 2, h = t & 3;
    float s = 0.f;
    for (int d = 0; d < HID; ++d)
        s += W_edge[k * FDIM + h * HID + d] * att_edge[h * HID + d];
    v[k * 4 + h] = s;
}

// node0[b*N+n] = x[b,0,n]   (feature 0, used for edge "diff")
__global__ void k_node0(const float* __restrict__ X, float* __restrict__ node0) {
    int i = blockIdx.x * blockDim.x + threadIdx.x;
    if (i >= BNTOT) return;
    int b = i >> 12, n = i & (NNODE - 1);
    node0[i] = X[b * (IN_DIM * NNODE) + n];
}

// ---------------------------------------------------------------------------
// Encoder: h = silu(x @ W1 + b1) @ W2 + b2 ; one 64-thread block per node.
__global__ void k_encoder(const float* __restrict__ X,
                          const float* __restrict__ W1, const float* __restrict__ b1,
                          const float* __restrict__ W2, const float* __restrict__ b2,
                          float* __restrict__ H) {
    const int node = blockIdx.x;
    const int d = threadIdx.x;              // 0..63
    const int b = node >> 12, n = node & (NNODE - 1);
    __shared__ float xin[IN_DIM];
    __shared__ float h1[HID];
    if (d < IN_DIM) xin[d] = X[b * (IN_DIM * NNODE) + d * NNODE + n];
    __syncthreads();
    float s = b1[d];
    #pragma unroll
    for (int c = 0; c < IN_DIM; ++c) s += xin[c] * W1[c * HID + d];
    h1[d] = s / (1.f + __expf(-s));         // silu
    __syncthreads();
    float t = b2[d];
    for (int k = 0; k < HID; ++k) t += h1[k] * W2[k * HID + d];
    H[node * HID + d] = t;
}

// ---------------------------------------------------------------------------
// xl = h @ W_gat  via V_WMMA_F32_16X16X4_F32.
// grid = (256, 16) blocks of 256 threads (8 waves); one 16x16 tile / wave.
// W tile (64 K-rows x 16 N-cols, 4 KB) staged global->LDS with the gfx1250
// async-to-LDS pipe (ASYNCcnt) when available.
__global__ void k_gemm_xl(const float* __restrict__ Hm,
                          const float* __restrict__ Wg,
                          float* __restrict__ XL) {
    __shared__ float lw[HID * 16];
    const int ntile = blockIdx.y;
    const int ncol0 = ntile * 16;

#if HAVE_ASYNC_LDS
    {
        // 1024 floats total: each of the 256 threads issues one B128 copy.
        // i = 4*tid -> row kk = i>>4, col nn = i&15 (nn in {0,4,8,12}), so the
        // 4 consecutive floats stay inside one 16-wide LDS row and one global row.
        const int i  = threadIdx.x * 4;
        const int kk = i >> 4, nn = i & 15;
        __builtin_amdgcn_global_load_async_to_lds_b128(
            (gv4i*)(Wg + kk * FDIM + ncol0 + nn),
            (lv4i*)&lw[i], 0, 0);
    }
    __builtin_amdgcn_s_wait_asynccnt(0);
    __syncthreads();
#else
    for (int i = threadIdx.x; i < HID * 16; i += 256) {
        int kk = i >> 4, nn = i & 15;
        lw[i] = Wg[kk * FDIM + ncol0 + nn];
    }
    __syncthreads();
#endif

    const int wave  = threadIdx.x >> 5;
    const int lane  = threadIdx.x & 31;
    const int mtile = blockIdx.x * 8 + wave;
    const int m     = lane & 15;
    const int khalf = (lane >> 4) << 1;     // 0 (lanes 0-15) or 2 (lanes 16-31)
    const int nn    = lane & 15;

    const float* arow = Hm + (mtile * 16 + m) * HID;
    __builtin_prefetch(arow, 0, 3);         // global_prefetch_b8 on the A stream

    v8f acc = {};
    #pragma unroll
    for (int k = 0; k < HID; k += 4) {
        v2f a, b;
        a.x = arow[k + khalf];
        a.y = arow[k + khalf + 1];
        b.x = lw[(k + khalf) * 16 + nn];
        b.y = lw[(k + khalf + 1) * 16 + nn];
        acc = __builtin_amdgcn_wmma_f32_16x16x4_f32(
            false, a, false, b, (short)0, acc, false, false);
    }
    // C/D layout: VGPR r -> M=r (lanes 0-15), M=r+8 (lanes 16-31)
    const int rowBase = mtile * 16 + ((lane >> 4) << 3);
    float* outp = XL + (size_t)rowBase * FDIM + ncol0 + nn;
    #pragma unroll
    for (int r = 0; r < 8; ++r) outp[(size_t)r * FDIM] = acc[r];
}

// ---------------------------------------------------------------------------
// a_s / a_d: one wave per (node, head); 64-dot via shuffle reduction.
__global__ void k_att_dots(const float* __restrict__ XL,
                           const float* __restrict__ att_src,
                           const float* __restrict__ att_dst,
                           float* __restrict__ ASD) {
    const int wid  = (blockIdx.x * blockDim.x + threadIdx.x) >> 5;
    const int lane = threadIdx.x & 31;
    const int node = wid >> 2, head = wid & 3;
    const float* xp = XL + (size_t)node * FDIM + head * HID;
    const float* as = att_src + head * HID;
    const float* ad = att_dst + head * HID;
    float ss = xp[lane] * as[lane] + xp[lane + 32] * as[lane + 32];
    float sd = xp[lane] * ad[lane] + xp[lane + 32] * ad[lane + 32];
    #pragma unroll
    for (int off = 16; off > 0; off >>= 1) {
        ss += __shfl_xor(ss, off, 32);
        sd += __shfl_xor(sd, off, 32);
    }
    if (lane == 0) {
        ASD[node * 8 + head]     = ss;
        ASD[node * 8 + 4 + head] = sd;
    }
}

// ---------------------------------------------------------------------------
// Structural per-dst softmax aggregation + mean-over-heads + LN + SiLU.
// One 64-thread block per destination node; <=8 structural neighbors.
__global__ void k_aggregate(const float* __restrict__ XL,
                            const float* __restrict__ ASD,
                            const float* __restrict__ node0,
                            const float* __restrict__ vedge,   // [3][4]
                            const float* __restrict__ b_gat,
                            const float* __restrict__ g_f,
                            const float* __restrict__ bt_f,
                            float* __restrict__ Kout) {
    const int node = blockIdx.x;
    const int d = threadIdx.x;                         // 0..63
    const int b = node >> 12, n = node & (NNODE - 1);
    const int r = n >> 6, c = n & 63;
    __shared__ float red[HID];

    int   srcIdx[8];
    float ux[8], uy[8];
    int   nv = 0;
    #pragma unroll
    for (int t = 0; t < 9; ++t) {
        if (t == 4) continue;
        int dy = t / 3 - 1, dx = t % 3 - 1;
        int rs = r - dy, cs = c - dx;
        if (rs < 0 || rs >= GRID_W || cs < 0 || cs >= GRID_W) continue;
        float inv = rsqrtf((float)(dx * dx + dy * dy));
        srcIdx[nv] = b * NNODE + rs * GRID_W + cs;
        ux[nv] = dx * inv;
        uy[nv] = dy * inv;
        ++nv;
    }

    const float n0d = node0[node];
    float outv = 0.f;
    for (int h = 0; h < HEADS; ++h) {
        const float adst = ASD[node * 8 + 4 + h];
        const float v0 = vedge[0 * 4 + h], v1 = vedge[1 * 4 + h], v2 = vedge[2 * 4 + h];
        float aval[8], amax = -3.0e38f;
        for (int e = 0; e < nv; ++e) {
            float diff = n0d - node0[srcIdx[e]];
            float a = ASD[srcIdx[e] * 8 + h] + adst + ux[e] * v0 + uy[e] * v1 + diff * v2;
            a = (a > 0.f) ? a : 0.2f * a;              // leaky_relu(0.2)
            aval[e] = a;
            amax = fmaxf(amax, a);
        }
        float z = 0.f, msg = 0.f;
        for (int e = 0; e < nv; ++e) {
            float p = __expf(aval[e] - amax);
            z += p;
            msg += p * XL[(size_t)srcIdx[e] * FDIM + h * HID + d];
        }
        outv += msg / (z + 1e-16f);
    }
    outv = outv * (1.f / HEADS) + b_gat[d];

    // LayerNorm over 64 dims
    red[d] = outv; __syncthreads();
    for (int off = 32; off > 0; off >>= 1) { if (d < off) red[d] += red[d + off]; __syncthreads(); }
    float mean = red[0] * (1.f / HID); __syncthreads();
    float dv = outv - mean;
    red[d] = dv * dv; __syncthreads();
    for (int off = 32; off > 0; off >>= 1) { if (d < off) red[d] += red[d + off]; __syncthreads(); }
    float var = red[0] * (1.f / HID);
    float t = dv * rsqrtf(var + 1e-5f) * g_f[d] + bt_f[d];
    Kout[(size_t)node * HID + d] = t / (1.f + __expf(-t));   // silu
}

// ---------------------------------------------------------------------------
// RK4 bookkeeping. mode 0: acc=wacc*k, htmp=h+hstep*k
//                  mode 1: acc+=wacc*k, htmp=h+hstep*k
//                  mode 2: h += DT/6*(acc+k)
__global__ void k_rk_combine(float* __restrict__ h, const float* __restrict__ k,
                             float* __restrict__ acc, float* __restrict__ htmp,
                             float wacc, float hstep, int mode) {
    int i = blockIdx.x * 256 + threadIdx.x;
    if (i >= BNTOT * HID) return;
    if (mode == 2) {
        h[i] += (DT / 6.f) * (acc[i] + k[i]);
    } else {
        float a = (mode == 0 ? 0.f : acc[i]) + wacc * k[i];
        acc[i] = a;
        htmp[i] = h[i] + hstep * k[i];
    }
}

// ---------------------------------------------------------------------------
// Decoder: z = silu(LN(h)@W_h1+b_h1); logit = z@W_h2+b_h2; mask with init_fire.
__global__ void k_decoder(const float* __restrict__ H,
                          const float* __restrict__ g_h, const float* __restrict__ bt_h,
                          const float* __restrict__ W1, const float* __restrict__ b1,
                          const float* __restrict__ W2, const float* __restrict__ b2,
                          const float* __restrict__ X, float* __restrict__ out) {
    const int node = blockIdx.x;
    const int d = threadIdx.x;
    const int b = node >> 12, n = node & (NNODE - 1);
    __shared__ float tln[HID];
    __shared__ float zsh[HID];
    __shared__ float red[HID];

    float hv = H[(size_t)node * HID + d];
    red[d] = hv; __syncthreads();
    for (int off = 32; off > 0; off >>= 1) { if (d < off) red[d] += red[d + off]; __syncthreads(); }
    float mean = red[0] * (1.f / HID); __syncthreads();
    float dv = hv - mean;
    red[d] = dv * dv; __syncthreads();
    for (int off = 32; off > 0; off >>= 1) { if (d < off) red[d] += red[d + off]; __syncthreads(); }
    float var = red[0] * (1.f / HID);
    tln[d] = dv * rsqrtf(var + 1e-5f) * g_h[d] + bt_h[d];
    __syncthreads();

    float s = b1[d];
    for (int k = 0; k < HID; ++k) s += tln[k] * W1[k * HID + d];
    zsh[d] = s / (1.f + __expf(-s));
    __syncthreads();

    red[d] = zsh[d] * W2[d]; __syncthreads();
    for (int off = 32; off > 0; off >>= 1) { if (d < off) red[d] += red[d + off]; __syncthreads(); }
    if (d == 0) {
        float logit = red[0] + b2[0];
        float fire = X[b * (IN_DIM * NNODE) + n];    // x[:,0] channel
        if (fire > 0.5f) logit = fmaxf(logit, 6.f);
        out[b * NNODE + n] = logit;
    }
}

// ---------------------------------------------------------------------------
extern "C" void kernel_launch(void* const* d_in, const int* in_sizes, int n_in,
                              void* d_out, int out_size, void* d_ws, size_t ws_size,
                              hipStream_t stream) {
    const float* x        = (const float*)d_in[0];
    // d_in[1] edge_index, d_in[2] edge_dirs: structural, recomputed on device
    const float* W_enc1   = (const float*)d_in[3];
    const float* b_enc1   = (const float*)d_in[4];
    const float* W_enc2   = (const float*)d_in[5];
    const float* b_enc2   = (const float*)d_in[6];
    const float* W_gat    = (const float*)d_in[7];
    const float* att_src  = (const float*)d_in[8];
    const float* att_dst  = (const float*)d_in[9];
    const float* W_edge   = (const float*)d_in[10];
    const float* att_edge = (const float*)d_in[11];
    const float* b_gat    = (const float*)d_in[12];
    const float* g_f      = (const float*)d_in[13];
    const float* bt_f     = (const float*)d_in[14];
    const float* g_h      = (const float*)d_in[15];
    const float* bt_h     = (const float*)d_in[16];
    const float* W_h1     = (const float*)d_in[17];
    const float* b_h1     = (const float*)d_in[18];
    const float* W_h2     = (const float*)d_in[19];
    const float* b_h2     = (const float*)d_in[20];
    float* outp = (float*)d_out;

    float* ws = (float*)d_ws;
    float* h     = ws;                       ws += (size_t)BNTOT * HID;
    float* htmp  = ws;                       ws += (size_t)BNTOT * HID;
    float* acc   = ws;                       ws += (size_t)BNTOT * HID;
    float* kcur  = ws;                       ws += (size_t)BNTOT * HID;
    float* xl    = ws;                       ws += (size_t)BNTOT * FDIM;
    float* asd   = ws;                       ws += (size_t)BNTOT * 8;
    float* node0 = ws;                       ws += (size_t)BNTOT;
    float* vedge = ws;                       ws += 16;

    k_edge_v<<<1, 32, 0, stream>>>(W_edge, att_edge, vedge);
    k_node0<<<BNTOT / 256, 256, 0, stream>>>(x, node0);
    k_encoder<<<BNTOT, HID, 0, stream>>>(x, W_enc1, b_enc1, W_enc2, b_enc2, h);

    auto f_eval = [&](const float* hin) {
        k_gemm_xl<<<dim3(256, 16), 256, 0, stream>>>(hin, W_gat, xl);
        k_att_dots<<<(BNTOT * HEADS) / 8, 256, 0, stream>>>(xl, att_src, att_dst, asd);
        k_aggregate<<<BNTOT, HID, 0, stream>>>(xl, asd, node0, vedge, b_gat, g_f, bt_f, kcur);
    };

    const int nelem_blocks = (BNTOT * HID) / 256;
    for (int step = 0; step < 4; ++step) {
        f_eval(h);
        k_rk_combine<<<nelem_blocks, 256, 0, stream>>>(h, kcur, acc, htmp, 1.f, 0.5f * DT, 0);
        f_eval(htmp);
        k_rk_combine<<<nelem_blocks, 256, 0, stream>>>(h, kcur, acc, htmp, 2.f, 0.5f * DT, 1);
        f_eval(htmp);
        k_rk_combine<<<nelem_blocks, 256, 0, stream>>>(h, kcur, acc, htmp, 2.f, DT, 1);
        f_eval(htmp);
        k_rk_combine<<<nelem_blocks, 256, 0, stream>>>(h, kcur, acc, htmp, 0.f, 0.f, 2);
    }

    k_decoder<<<BNTOT, HID, 0, stream>>>(h, g_h, bt_h, W_h1, b_h1, W_h2, b_h2, x, outp);
}